// CascadedQuantization_39092792328249
// MI455X (gfx1250) — compile-verified
//
#include <hip/hip_runtime.h>

typedef float v2f __attribute__((ext_vector_type(2)));
typedef float v8f __attribute__((ext_vector_type(8)));

// Problem constants (from reference)
constexpr int NB = 8;           // batch
constexpr int NC = 256;         // channels
constexpr int NSP = 32 * 32;    // spatial N = H*W = 1024
constexpr int NL = 4;           // levels
constexpr int NG = 4;           // groups
constexpr int NK = 2048;        // codewords per group
constexpr int ND = 64;          // dims per group (C/G)

constexpr int ROWS = 16;        // rows (feature vectors) per block = one WMMA M-tile
constexpr int RSTR = 68;        // padded LDS row stride (floats) -> conflict-free A reads

// ---- precompute 0.5*||c||^2 for every codeword: [L*G*K] floats (128 KB in d_ws) ----
__global__ __launch_bounds__(256)
void rvq_cnorm_kernel(const float* __restrict__ cb, float* __restrict__ cnorm) {
    int i = blockIdx.x * blockDim.x + threadIdx.x;       // 0 .. L*G*K-1
    const float* row = cb + (size_t)i * ND;
    float s = 0.f;
    #pragma unroll 8
    for (int d = 0; d < ND; ++d) { float v = row[d]; s += v * v; }
    cnorm[i] = 0.5f * s;
}

// ---- main cascaded-VQ kernel: one wave per 16-row tile of one group ----
__global__ __launch_bounds__(32)
void rvq_main_kernel(const float* __restrict__ x,
                     const float* __restrict__ codebooks,
                     const float* __restrict__ cnorm,
                     float* __restrict__ out) {
    __shared__ float rT[ROWS * RSTR];   // residual tile [16][64] (padded)
    __shared__ float xT[ROWS * RSTR];   // xhat tile
    __shared__ int   kRow[ROWS];        // winning codeword per row

    const int lane = threadIdx.x;                  // 0..31 (wave32)
    const int g    = blockIdx.x >> 9;              // 4 groups x 512 tiles
    const int tile = blockIdx.x & 511;
    const int row0 = tile * ROWS;                  // row within [0, B*N)
    const int b    = row0 >> 10;                   // / 1024
    const int nsp  = row0 & 1023;                  // spatial base index

    // x flat index: ((b*NC + g*ND + d) * NSP + n)
    const size_t xbase = ((size_t)b * NC + (size_t)g * ND) * NSP + nsp;

    // Load residual tile (coalesced: m fastest), zero xhat tile.
    for (int i = lane; i < ROWS * ND; i += 32) {
        int m = i & 15;
        int d = i >> 4;
        rT[m * RSTR + d] = x[xbase + (size_t)d * NSP + m];
        xT[m * RSTR + d] = 0.f;
    }
    __syncthreads();

    const int M    = lane & 15;                    // A-matrix row for this lane
    const int koff = (lane >> 4) << 1;             // K sub-offset: 0 (lanes 0-15) / 2 (16-31)
    const int col  = lane & 15;                    // B/C column for this lane

    for (int l = 0; l < NL; ++l) {
        const float* cbg = codebooks + (size_t)(l * NG + g) * NK * ND;
        const float* cng = cnorm     + (size_t)(l * NG + g) * NK;

        // A operand regs: 16 K-slices of 4, per ISA 16x4 f32 A layout.
        v2f a[16];
        #pragma unroll
        for (int ks = 0; ks < 16; ++ks)
            a[ks] = *(const v2f*)&rT[M * RSTR + ks * 4 + koff];

        float bestS[8];
        int   bestI[8];
        #pragma unroll
        for (int v = 0; v < 8; ++v) { bestS[v] = -3.402823466e38f; bestI[v] = 0; }

        // Scan all 2048 codewords in 128 tiles of 16 columns.
        for (int kt = 0; kt < NK / 16; ++kt) {
            const int n = kt * 16 + col;           // codeword index for this lane's column
            const float bias = -cng[n];            // fold -0.5*||c||^2 into the accumulator
            v8f acc;
            #pragma unroll
            for (int v = 0; v < 8; ++v) acc[v] = bias;

            const float* bp = cbg + (size_t)n * ND + koff;  // B operand: column n, this half's K rows
            #pragma unroll
            for (int ks = 0; ks < 16; ++ks) {
                v2f bv = *(const v2f*)(bp + ks * 4);
                acc = __builtin_amdgcn_wmma_f32_16x16x4_f32(
                    false, a[ks], false, bv, (short)0, acc, false, false);
            }

            // Running argmax per (row-slot, column-residue); strict > keeps first occurrence.
            #pragma unroll
            for (int v = 0; v < 8; ++v) {
                float s = acc[v];
                if (s > bestS[v]) { bestS[v] = s; bestI[v] = n; }
            }
        }

        // Reduce across the 16 lanes of each half (masks 1..8 stay within a half on wave32).
        #pragma unroll
        for (int v = 0; v < 8; ++v) {
            float s = bestS[v]; int idx = bestI[v];
            #pragma unroll
            for (int off = 8; off >= 1; off >>= 1) {
                float so = __shfl_xor(s, off, 32);
                int   io = __shfl_xor(idx, off, 32);
                if (so > s || (so == s && io < idx)) { s = so; idx = io; }
            }
            bestS[v] = s; bestI[v] = idx;
        }
        if (lane == 0) {
            #pragma unroll
            for (int v = 0; v < 8; ++v) kRow[v] = bestI[v];       // rows 0..7
        }
        if (lane == 16) {
            #pragma unroll
            for (int v = 0; v < 8; ++v) kRow[v + 8] = bestI[v];   // rows 8..15
        }
        __syncthreads();

        // r -= q ; xhat += q   (q = winning codeword row, coalesced reads)
        for (int i = lane; i < ROWS * ND; i += 32) {
            int m = i >> 6;
            int d = i & 63;
            float q = cbg[(size_t)kRow[m] * ND + d];
            rT[m * RSTR + d] -= q;
            xT[m * RSTR + d] += q;
        }
        __syncthreads();
    }

    // Write xhat back in x layout.
    for (int i = lane; i < ROWS * ND; i += 32) {
        int m = i & 15;
        int d = i >> 4;
        out[xbase + (size_t)d * NSP + m] = xT[m * RSTR + d];
    }
}

extern "C" void kernel_launch(void* const* d_in, const int* in_sizes, int n_in,
                              void* d_out, int out_size, void* d_ws, size_t ws_size,
                              hipStream_t stream) {
    const float* x   = (const float*)d_in[0];       // [8,256,32,32]
    const float* cbs = (const float*)d_in[1];       // [4,4,2048,64]
    float* out   = (float*)d_out;                   // [8,256,32,32]
    float* cnorm = (float*)d_ws;                    // L*G*K floats = 128 KB

    rvq_cnorm_kernel<<<(NL * NG * NK) / 256, 256, 0, stream>>>(cbs, cnorm);

    const int tilesPerGroup = (NB * NSP) / ROWS;    // 512
    rvq_main_kernel<<<NG * tilesPerGroup, 32, 0, stream>>>(x, cbs, cnorm, out);
}